// CrfDecodeLayer_10256381903473
// MI455X (gfx1250) — compile-verified
//
#include <hip/hip_runtime.h>
#include <stdint.h>

#define BB 256
#define TT 512
#define KK 256
#define OUTW 514  // MAX_SEQ + 2

typedef __attribute__((ext_vector_type(4))) unsigned int u32x4;
typedef __attribute__((ext_vector_type(8))) int i32x8;
typedef __attribute__((ext_vector_type(4))) int i32x4;

__global__ __launch_bounds__(KK) void crf_decode_kernel(
    const float* __restrict__ em,     // [B, T, K]
    const float* __restrict__ tr,     // [K, K]
    const int*   __restrict__ mask,   // [B, T]
    float*       __restrict__ out,    // [B, OUTW]
    uint8_t*     __restrict__ bp)     // [B, T, K] scratch backpointers
{
    extern __shared__ float smem[];
    float* s_trans = smem;            // KK*KK floats (256 KB)
    float* s_state = smem + KK * KK;  // KK floats (16B-aligned: 256KB offset)

    const int b = blockIdx.x;
    const int j = threadIdx.x;

    // ---- Stage transitions matrix into LDS via the Tensor Data Mover ----
#if defined(__gfx1250__) && __has_builtin(__builtin_amdgcn_tensor_load_to_lds)
    if (j < 32) {  // wave 0 issues one TDM op moving the whole 256x256 f32 tile
        const uint32_t lds_off = (uint32_t)(uintptr_t)(void*)s_trans; // low 32b of flat LDS addr = LDS offset
        const uint64_t ga = (uint64_t)(uintptr_t)tr;
        u32x4 g0;
        g0[0] = 1u;                                   // count=1, user descriptor, no gather
        g0[1] = lds_off;                              // lds_addr
        g0[2] = (uint32_t)(ga & 0xFFFFFFFFu);         // global_addr[31:0]
        g0[3] = ((uint32_t)(ga >> 32) & 0x01FFFFFFu)  // global_addr[56:32]
                | 0x80000000u;                        // type = 2 ("image")
        i32x8 g1;
        g1[0] = 0x00020000;         // workgroup_mask=0, data_size=2 (4 bytes)
        g1[1] = (int)(KK << 16);    // tensor_dim0[15:0] in bits[63:48]
        g1[2] = (int)(KK << 16);    // tensor_dim1[15:0] in bits[95:80]
        g1[3] = (int)(KK << 16);    // tile_dim0 in bits[127:112]
        g1[4] = (int)KK;            // tile_dim1 in bits[143:128]
        g1[5] = (int)KK;            // tensor_dim0_stride[31:0]
        g1[6] = 0;                  // stride0 hi / stride1 lo (unused for 2D)
        g1[7] = 0;
        i32x4 z4 = {0, 0, 0, 0};             // groups 2/3 unused (2D tensor)
        i32x8 z8 = {0, 0, 0, 0, 0, 0, 0, 0}; // unused trailing group
        __builtin_amdgcn_tensor_load_to_lds(g0, g1, z4, z4, z8, 0);
        __builtin_amdgcn_s_wait_tensorcnt(0);
    }
#else
    for (int r = 0; r < KK; ++r)
        s_trans[r * KK + j] = tr[r * KK + j];
#endif

    // ---- Init state with t=0 emissions ----
    float st = em[((size_t)b * TT + 0) * KK + j];
    s_state[j] = st;
    __syncthreads();

    const int* mrow = mask + (size_t)b * TT;

    // ---- Forward Viterbi: 511 sequential max-plus GEMV steps ----
    float em_cur = em[((size_t)b * TT + 1) * KK + j];
    for (int t = 1; t < TT; ++t) {
        // software-pipelined emission prefetch for next step
        float em_next = (t + 1 < TT) ? em[((size_t)b * TT + (t + 1)) * KK + j] : 0.0f;

        // 4 independent argmax streams over contiguous i-blocks (ILP), with
        // state broadcasts widened to ds_load_b128 (4 states per DS op).
        float b0 = -__builtin_inff(), b1 = -__builtin_inff();
        float b2 = -__builtin_inff(), b3 = -__builtin_inff();
        int a0 = 0, a1 = 64, a2 = 128, a3 = 192;
#pragma unroll 4
        for (int i = 0; i < 64; i += 4) {
            const float4 s0 = *(const float4*)(s_state + i      );
            const float4 s1 = *(const float4*)(s_state + i + 64 );
            const float4 s2 = *(const float4*)(s_state + i + 128);
            const float4 s3 = *(const float4*)(s_state + i + 192);
            float v;
            v = s0.x + s_trans[(i +   0) * KK + j]; if (v > b0) { b0 = v; a0 = i +   0; }
            v = s0.y + s_trans[(i +   1) * KK + j]; if (v > b0) { b0 = v; a0 = i +   1; }
            v = s0.z + s_trans[(i +   2) * KK + j]; if (v > b0) { b0 = v; a0 = i +   2; }
            v = s0.w + s_trans[(i +   3) * KK + j]; if (v > b0) { b0 = v; a0 = i +   3; }
            v = s1.x + s_trans[(i +  64) * KK + j]; if (v > b1) { b1 = v; a1 = i +  64; }
            v = s1.y + s_trans[(i +  65) * KK + j]; if (v > b1) { b1 = v; a1 = i +  65; }
            v = s1.z + s_trans[(i +  66) * KK + j]; if (v > b1) { b1 = v; a1 = i +  66; }
            v = s1.w + s_trans[(i +  67) * KK + j]; if (v > b1) { b1 = v; a1 = i +  67; }
            v = s2.x + s_trans[(i + 128) * KK + j]; if (v > b2) { b2 = v; a2 = i + 128; }
            v = s2.y + s_trans[(i + 129) * KK + j]; if (v > b2) { b2 = v; a2 = i + 129; }
            v = s2.z + s_trans[(i + 130) * KK + j]; if (v > b2) { b2 = v; a2 = i + 130; }
            v = s2.w + s_trans[(i + 131) * KK + j]; if (v > b2) { b2 = v; a2 = i + 131; }
            v = s3.x + s_trans[(i + 192) * KK + j]; if (v > b3) { b3 = v; a3 = i + 192; }
            v = s3.y + s_trans[(i + 193) * KK + j]; if (v > b3) { b3 = v; a3 = i + 193; }
            v = s3.z + s_trans[(i + 194) * KK + j]; if (v > b3) { b3 = v; a3 = i + 194; }
            v = s3.w + s_trans[(i + 195) * KK + j]; if (v > b3) { b3 = v; a3 = i + 195; }
        }
        // merge: strict '>' with ascending block order == jnp.argmax first-max
        float bestv = b0; int besti = a0;
        if (b1 > bestv) { bestv = b1; besti = a1; }
        if (b2 > bestv) { bestv = b2; besti = a2; }
        if (b3 > bestv) { bestv = b3; besti = a3; }

        // backpointer as u8 (K=256 fits a byte): 4x less scratch traffic
        bp[((size_t)b * TT + t) * KK + j] = (uint8_t)besti;

        float ns = bestv + em_cur;
        int mk = mrow[t];
        float newst = (mk > 0) ? ns : st;

        __syncthreads();          // all reads of old state complete
        s_state[j] = newst;
        st = newst;
        __syncthreads();          // new state visible

        em_cur = em_next;
    }

    // make backpointer stores visible within the workgroup
    __threadfence_block();
    __syncthreads();

    // ---- Backward pass + output (thread 0 per batch) ----
    if (j == 0) {
        // argmax of final state (first occurrence of max)
        float bv = s_state[0]; int tag = 0;
        for (int q = 1; q < KK; ++q) {
            float v = s_state[q];
            if (v > bv) { bv = v; tag = q; }
        }
        float* orow = out + (size_t)b * OUTW;
        orow[OUTW - 1] = 0.0f;   // padding
        orow[OUTW - 2] = 0.0f;   // padding
        for (int t = TT - 1; t >= 1; --t) {
            int mk = mrow[t];
            orow[t] = (float)(tag * mk);                        // tags * mask
            int prev = (int)bp[((size_t)b * TT + t) * KK + tag];
            tag = (mk > 0) ? prev : tag;
        }
        orow[0] = (float)(tag * mrow[0]);
    }
}

extern "C" void kernel_launch(void* const* d_in, const int* in_sizes, int n_in,
                              void* d_out, int out_size, void* d_ws, size_t ws_size,
                              hipStream_t stream) {
    const float* em   = (const float*)d_in[0];   // emissions [B,T,K] f32
    const float* tr   = (const float*)d_in[1];   // transitions [K,K] f32
    const int*   mask = (const int*)d_in[2];     // mask [B,T] i32
    // d_in[3] = max_sequence_length (compile-time 512 here)
    float*   out = (float*)d_out;                // [B, 514]
    uint8_t* bp  = (uint8_t*)d_ws;               // [B,T,K] u8 backpointers (32 MB)

    const size_t shmem = (size_t)(KK * KK + KK) * sizeof(float);  // 257 KB (< 320 KB WGP LDS)
    crf_decode_kernel<<<BB, KK, shmem, stream>>>(em, tr, mask, out, bp);
}